// SOT_42992622633415
// MI455X (gfx1250) — compile-verified
//
#include <hip/hip_runtime.h>

#define NB 8
#define NP 1024
#define NNE (NP * NP)          // 1M elements per batch
#define REGP 0.1f
#define TOLC 1e-3f

typedef float v2f __attribute__((ext_vector_type(2)));
typedef float v8f __attribute__((ext_vector_type(8)));

// ---- CDNA5 f32 WMMA: D(16x16,f32) = A(16x4,f32) x B(4x16,f32) + C ----
__device__ __forceinline__ v8f wmma4f32(v2f a, v2f b, v8f c) {
  // 8-arg form: (neg_a, A, neg_b, B, c_mod, C, reuse_a, reuse_b)
  return __builtin_amdgcn_wmma_f32_16x16x4_f32(false, a, false, b, (short)0, c,
                                               false, false);
}

__device__ __forceinline__ float pick8(v8f c, int i) {
  switch (i & 7) {
    case 0: return c[0]; case 1: return c[1]; case 2: return c[2]; case 3: return c[3];
    case 4: return c[4]; case 5: return c[5]; case 6: return c[6]; default: return c[7];
  }
}

__device__ __forceinline__ float wave_max32(float v) {
#pragma unroll
  for (int off = 16; off > 0; off >>= 1) v = fmaxf(v, __shfl_xor(v, off, 32));
  return v;
}

// Stage 4KB (1024 floats) global -> LDS with CDNA5 async copies (ASYNCcnt path).
// Each of the 128 block threads issues two b128 transfers; LDS operand is the
// per-lane LDS byte offset (= low 32 bits of the generic shared pointer).
__device__ __forceinline__ void async_stage_1k(float* lds, const float* gsrc) {
  const int t = threadIdx.x;                 // 0..127
  unsigned lbase = (unsigned)(size_t)lds;
#pragma unroll
  for (int q = 0; q < 2; ++q) {
    unsigned laddr = lbase + (unsigned)(t + 128 * q) * 16u;
    const float* g = gsrc + (t + 128 * q) * 4;
    asm volatile("global_load_async_to_lds_b128 %0, %1, off"
                 :: "v"(laddr), "v"(g) : "memory");
  }
  asm volatile("s_wait_asynccnt 0x0" ::: "memory");
}

// ---------------- init scratch scalars ----------------
__global__ void sk_init(int* maxMbits, float* suma, float* sumb, int* maxdif, int* flags) {
  int t = threadIdx.x;
  if (t < NB) { maxMbits[t] = 0; suma[t] = 0.f; sumb[t] = 0.f; }
  if (t == 0) { maxdif[0] = 0; flags[0] = 0; flags[1] = 0; }
}

// ---------------- M[i,j] = |x_i|^2 + |y_j|^2 - 2 x_i.y_j  (WMMA rank-2 GEMM) ----------------
__global__ void sk_pairwise(const float* __restrict__ c1, const float* __restrict__ c2,
                            float* __restrict__ M, int* __restrict__ maxMbits) {
  const int lane = threadIdx.x;
  const int hi = lane >> 4, l15 = lane & 15;
  const int j0 = blockIdx.x * 16, i0 = blockIdx.y * 16, bz = blockIdx.z;
  const float2* X = (const float2*)(c1 + (size_t)bz * NP * 2);
  const float2* Y = (const float2*)(c2 + (size_t)bz * NP * 2);
  float2 x2 = X[i0 + l15];
  float2 y2 = Y[j0 + l15];
  // A: lane<16 holds row m=lane, k=0..1; lane>=16 holds k=2..3 (zero pad, D=2)
  v2f A, Bv;
  A[0]  = hi ? 0.f : x2.x;  A[1]  = hi ? 0.f : x2.y;
  Bv[0] = hi ? 0.f : y2.x;  Bv[1] = hi ? 0.f : y2.y;  // B[k,n]=y[n][k], rows 2,3 zero
  v8f acc = {};
  acc = wmma4f32(A, Bv, acc);              // acc[r] = x_{m}.y_{n}, m=r+8*hi, n=l15
  float yn = y2.x * y2.x + y2.y * y2.y;    // n = l15 for both halves
  float* Mb = M + (size_t)bz * NNE;
  float tmax = 0.f;
#pragma unroll
  for (int r = 0; r < 8; ++r) {
    float2 xr = X[i0 + 8 * hi + r];
    float xn = xr.x * xr.x + xr.y * xr.y;
    float mv = xn + yn - 2.f * acc[r];
    Mb[(size_t)(i0 + 8 * hi + r) * NP + j0 + l15] = mv;
    tmax = fmaxf(tmax, mv);
  }
  tmax = wave_max32(tmax);
  // int-compare atomicMax is correct for the (positive) true max; deterministic.
  if (lane == 0) atomicMax(&maxMbits[bz], __float_as_int(tmax));
}

// ---------------- per-batch sums of (prob + 1e-8) (deterministic tree) ----------------
__global__ void sk_sums(const float* __restrict__ p1, const float* __restrict__ p2,
                        float* suma, float* sumb) {
  __shared__ float ra[256], rb[256];
  const int bz = blockIdx.x, t = threadIdx.x;
  float sa = 0.f, sb = 0.f;
#pragma unroll
  for (int q = 0; q < NP; q += 256) {
    sa += p1[bz * NP + q + t] + 1e-8f;
    sb += p2[bz * NP + q + t] + 1e-8f;
  }
  ra[t] = sa; rb[t] = sb; __syncthreads();
  for (int s = 128; s > 0; s >>= 1) {
    if (t < s) { ra[t] += ra[t + s]; rb[t] += rb[t + s]; }
    __syncthreads();
  }
  if (t == 0) { suma[bz] = ra[0]; sumb[bz] = rb[0]; }
}

// ---------------- normalize a,b; init u,v = 1/N ----------------
__global__ void sk_norm(const float* __restrict__ p1, const float* __restrict__ p2,
                        const float* __restrict__ suma, const float* __restrict__ sumb,
                        float* a, float* b, float* u, float* v) {
  int i = blockIdx.x * blockDim.x + threadIdx.x;   // 8192
  int bz = i >> 10;
  a[i] = (p1[i] + 1e-8f) / suma[bz];
  b[i] = (p2[i] + 1e-8f) / sumb[bz];
  u[i] = 1.f / NP;
  v[i] = 1.f / NP;
}

// ---------------- K = exp(-M/(REG*maxM)); P0 = K/N^2  (b128 streaming) ----------------
__global__ void sk_expK(float* __restrict__ K, float* __restrict__ P,
                        const int* __restrict__ maxMbits) {
  size_t i4 = (size_t)blockIdx.x * blockDim.x + threadIdx.x;
  int bz = (int)(i4 >> 18);                        // (i4*4) >> 20
  float inv = 1.f / (REGP * __int_as_float(maxMbits[bz]));
  float4 m = ((const float4*)K)[i4];
  float4 kv;
  kv.x = __expf(-m.x * inv); kv.y = __expf(-m.y * inv);
  kv.z = __expf(-m.z * inv); kv.w = __expf(-m.w * inv);
  ((float4*)K)[i4] = kv;
  const float s = 1.f / ((float)NP * (float)NP);
  float4 p; p.x = kv.x * s; p.y = kv.y * s; p.z = kv.z * s; p.w = kv.w * s;
  ((float4*)P)[i4] = p;
}

// ---------------- v = b / (K^T u): 4 waves/block share LDS-staged u ----------------
__global__ void sk_colmv(const float* __restrict__ K, const float* __restrict__ u,
                         const float* __restrict__ b, float* __restrict__ v,
                         int* __restrict__ flags) {
  if (flags[0] | flags[1]) return;               // frozen (converged or numerical error)
  __shared__ float su[NP];
  const int bz = blockIdx.y;
  async_stage_1k(su, u + bz * NP);               // ASYNCcnt-tracked global->LDS copy
  __syncthreads();
  const int lane = threadIdx.x & 31, wid = threadIdx.x >> 5;
  const int j0 = (blockIdx.x * 4 + wid) * 16;
  const float* Kb = K + (size_t)bz * NNE;
  const int hi = lane >> 4, l15 = lane & 15, koff = hi * 2;
  v8f acc0 = {}, acc1 = {};
  for (int kk = 0; kk < NP; kk += 16) {
    __builtin_prefetch(&Kb[(size_t)((kk + 128) & (NP - 1)) * NP + j0 + l15], 0, 1);
#pragma unroll
    for (int c = 0; c < 4; ++c) {
      const int k4 = kk + 4 * c + koff;
      v2f A, Bv;
      // A[m,k] = K^T[j0+m, k4+k] = K[(k4+k)*N + j0+m]  (coalesced across lanes)
      A[0] = Kb[(size_t)k4 * NP + j0 + l15];
      A[1] = Kb[(size_t)(k4 + 1) * NP + j0 + l15];
      Bv[0] = su[k4];                            // broadcast u chunk: B[k,n]=u[k4+k]
      Bv[1] = su[k4 + 1];
      if (c & 1) acc1 = wmma4f32(A, Bv, acc1);
      else       acc0 = wmma4f32(A, Bv, acc0);
    }
  }
  if (l15 < 8) {                                 // lanes 0-7 -> m=0..7, lanes 16-23 -> m=8..15
    int m = l15 + hi * 8;
    float ktu = pick8(acc0, l15) + pick8(acc1, l15);
    float v1 = b[bz * NP + j0 + m] / ktu;
    if (ktu == 0.f || !(fabsf(v1) < __builtin_inff())) atomicOr(&flags[1], 1);
    v[bz * NP + j0 + m] = v1;
  }
}

// ---------------- u = a / (K v): 4 waves/block share LDS-staged v ----------------
__global__ void sk_rowmv(const float* __restrict__ K, const float* __restrict__ v,
                         const float* __restrict__ a, float* __restrict__ u,
                         int* __restrict__ flags) {
  if (flags[0] | flags[1]) return;
  __shared__ float sv[NP];
  const int bz = blockIdx.y;
  async_stage_1k(sv, v + bz * NP);
  __syncthreads();
  const int lane = threadIdx.x & 31, wid = threadIdx.x >> 5;
  const int i0 = (blockIdx.x * 4 + wid) * 16;
  const float* Kb = K + (size_t)bz * NNE;
  const int hi = lane >> 4, l15 = lane & 15, koff = hi * 2;
  const size_t rowb = (size_t)(i0 + l15) * NP;
  v8f acc0 = {}, acc1 = {};
  for (int kk = 0; kk < NP; kk += 16) {
    __builtin_prefetch(&Kb[rowb + ((kk + 128) & (NP - 1))], 0, 1);
#pragma unroll
    for (int c = 0; c < 4; ++c) {
      const int k4 = kk + 4 * c + koff;
      // A[m,k] = K[(i0+m)*N + k4+k]: contiguous b64 per lane from its row
      const float2 kf = *(const float2*)&Kb[rowb + k4];
      v2f A;  A[0] = kf.x;  A[1] = kf.y;
      v2f Bv; Bv[0] = sv[k4]; Bv[1] = sv[k4 + 1];
      if (c & 1) acc1 = wmma4f32(A, Bv, acc1);
      else       acc0 = wmma4f32(A, Bv, acc0);
    }
  }
  if (l15 < 8) {
    int m = l15 + hi * 8;
    float kv = pick8(acc0, l15) + pick8(acc1, l15);
    float u1 = a[bz * NP + i0 + m] / kv;
    if (!(fabsf(u1) < __builtin_inff())) atomicOr(&flags[1], 1);
    u[bz * NP + i0 + m] = u1;
  }
}

// ---------------- Pn = u*K*v^T; maxdiff = max|P - Pn|; P <- Pn  (b128 streaming) ------------
__global__ void sk_plan(const float* __restrict__ K, const float* __restrict__ u,
                        const float* __restrict__ v, float* __restrict__ P,
                        int* __restrict__ maxdif, const int* __restrict__ flags) {
  if (flags[0] | flags[1]) return;
  __shared__ float red[256];
  const int t = threadIdx.x;
  size_t i4 = (size_t)blockIdx.x * 256 + t;
  size_t i = i4 * 4;
  int bz = (int)(i >> 20);
  int row = (int)(i >> 10) & (NP - 1);
  int col = (int)i & (NP - 1);
  float uu = u[bz * NP + row];
  float4 k4 = ((const float4*)K)[i4];
  float4 vv = *(const float4*)&v[bz * NP + col];
  float4 po = ((const float4*)P)[i4];
  float4 pn;
  pn.x = uu * k4.x * vv.x; pn.y = uu * k4.y * vv.y;
  pn.z = uu * k4.z * vv.z; pn.w = uu * k4.w * vv.w;
  float lm = fmaxf(fmaxf(fabsf(po.x - pn.x), fabsf(po.y - pn.y)),
                   fmaxf(fabsf(po.z - pn.z), fabsf(po.w - pn.w)));
  ((float4*)P)[i4] = pn;
  red[t] = lm; __syncthreads();
  for (int s = 128; s > 0; s >>= 1) {
    if (t < s) red[t] = fmaxf(red[t], red[t + s]);
    __syncthreads();
  }
  if (t == 0) atomicMax(maxdif, __float_as_int(red[0]));   // nonneg -> int-max OK
}

__global__ void sk_flag(int* maxdif, int* flags) {
  if (!(flags[0] | flags[1]) && __int_as_float(maxdif[0]) < TOLC) flags[0] = 1;
  maxdif[0] = 0;
}

extern "C" void kernel_launch(void* const* d_in, const int* in_sizes, int n_in,
                              void* d_out, int out_size, void* d_ws, size_t ws_size,
                              hipStream_t stream) {
  const float* c1 = (const float*)d_in[0];   // coord1 [8,1024,2]
  const float* p1 = (const float*)d_in[1];   // prob1  [8,1024]
  const float* c2 = (const float*)d_in[2];   // coord2 [8,1024,2]
  const float* p2 = (const float*)d_in[3];   // prob2  [8,1024]
  float* P = (float*)d_out;                  // [8,1024,1024]

  float* ws = (float*)d_ws;
  float* K = ws;                                   // 8M floats (32 MB, lives in L2)
  float* a = ws + (size_t)NB * NNE;                // 8192
  float* b = a + NB * NP;
  float* u = b + NB * NP;
  float* v = u + NB * NP;
  int*   maxMbits = (int*)(v + NB * NP);           // NB
  float* suma = (float*)(maxMbits + NB);           // NB
  float* sumb = suma + NB;                         // NB
  int*   maxdif = (int*)(sumb + NB);               // 1
  int*   flags  = maxdif + 1;                      // [done, bad]

  sk_init<<<1, 64, 0, stream>>>(maxMbits, suma, sumb, maxdif, flags);
  sk_pairwise<<<dim3(NP / 16, NP / 16, NB), 32, 0, stream>>>(c1, c2, K, maxMbits);
  sk_sums<<<NB, 256, 0, stream>>>(p1, p2, suma, sumb);
  sk_norm<<<NB * NP / 256, 256, 0, stream>>>(p1, p2, suma, sumb, a, b, u, v);
  sk_expK<<<(int)((size_t)NB * NNE / 1024), 256, 0, stream>>>(K, P, maxMbits);

  for (int o = 0; o < 10; ++o) {                   // MAX_ITERS/INNER outer steps
    for (int it = 0; it < 10; ++it) {              // INNER
      sk_colmv<<<dim3(NP / 64, NB), 128, 0, stream>>>(K, u, b, v, flags);
      sk_rowmv<<<dim3(NP / 64, NB), 128, 0, stream>>>(K, v, a, u, flags);
    }
    sk_plan<<<(int)((size_t)NB * NNE / 1024), 256, 0, stream>>>(K, u, v, P, maxdif, flags);
    sk_flag<<<1, 1, 0, stream>>>(maxdif, flags);
  }
}